// GCNLayer_63513976373549
// MI455X (gfx1250) — compile-verified
//
#include <hip/hip_runtime.h>

// GCN layer: out = relu(segment_sum(h[src], dst) @ W^T)
//   h  : [100000, 64] f32     src/dst : [1250000] i64     W : [64, 64] f32
// Three kernels on `stream`:
//   1) zero agg (d_ws scratch, 25.6 MB)
//   2) edge-parallel gather + L2 atomic scatter-add (dominant cost, L2-resident)
//   3) WMMA f32 GEMM + ReLU  (v_wmma_f32_16x16x4_f32, exact fp32)

#define N_NODES 100000
#define N_EDGES 1250000
#define DFEAT   64

typedef __attribute__((ext_vector_type(2))) float v2f;
typedef __attribute__((ext_vector_type(8))) float v8f;

// ---------------------------------------------------------------- kernel 1
__global__ void gcn_zero_agg(float* __restrict__ agg, int n_elts) {
    int i = (blockIdx.x * blockDim.x + threadIdx.x) * 4;
    if (i < n_elts) {
        *(float4*)(agg + i) = make_float4(0.f, 0.f, 0.f, 0.f);
    }
}

// ---------------------------------------------------------------- kernel 2
// 16 threads per edge; each thread moves one float4 (16B) of the 256B row.
// Threads in a group all load the same src[e]/dst[e] (merged by coalescer).
// h (25.6MB) and agg (25.6MB) both fit in the 192MB L2, so the 320MB gather
// + 320MB atomic-scatter traffic is L2-resident after first touch.
__global__ void gcn_scatter_edges(const float* __restrict__ h,
                                  const long long* __restrict__ src,
                                  const long long* __restrict__ dst,
                                  float* __restrict__ agg) {
    long long t = (long long)blockIdx.x * blockDim.x + threadIdx.x;
    int e = (int)(t >> 4);          // edge index
    int c = (int)(t & 15);          // 16B chunk within the 64-float row
    if (e >= N_EDGES) return;

    int s = (int)src[e];
    int d = (int)dst[e];

    const float4 v = *(const float4*)(h + (long long)s * DFEAT + c * 4);
    float* p = agg + (long long)d * DFEAT + c * 4;
    // GLOBAL_ATOMIC_ADD_F32, no return -> STOREcnt only, resolved in L2
    atomicAdd(p + 0, v.x);
    atomicAdd(p + 1, v.y);
    atomicAdd(p + 2, v.z);
    atomicAdd(p + 3, v.w);
}

// ---------------------------------------------------------------- kernel 3
// out[n,o] = relu( sum_k agg[n,k] * W[o,k] )
// One 16x16 output tile per wave; block = 4 waves = the 4 column tiles of
// one 16-row strip. K=64 -> 16 chained V_WMMA_F32_16X16X4_F32.
//
// Fragment layout (ISA 7.12.2, 32-bit):
//   A 16x4 : lane L (m=L%16, half=L/16), vgpr v -> A[m, k0 + 2*half + v]
//   B 4x16 : lane L, vgpr v               -> B[k0 + 2*half + v, m]
//   C 16x16: lane L, vgpr v               -> C[v + 8*half, m]
// B(k,n) = W[n*64 + k]  (W is [out,in] row-major), so both fragments are
// plain strided float2 loads from row-major storage.
__global__ void __launch_bounds__(128)
gcn_gemm_relu(const float* __restrict__ agg,
              const float* __restrict__ W,
              float* __restrict__ out) {
    const int wave    = threadIdx.x >> 5;   // 0..3 -> column tile
    const int lane    = threadIdx.x & 31;
    const int m       = lane & 15;
    const int half    = lane >> 4;
    const int rowbase = blockIdx.x << 4;    // 100000 % 16 == 0: EXEC all-1s
    const int colbase = wave << 4;

    const float* __restrict__ arow = agg + (long long)(rowbase + m) * DFEAT;
    const float* __restrict__ brow = W   + (long long)(colbase + m) * DFEAT;

    v8f c = {};
#pragma unroll
    for (int k0 = 0; k0 < DFEAT; k0 += 4) {
        const int k = k0 + 2 * half;
        v2f a, b;
        a.x = arow[k]; a.y = arow[k + 1];
        b.x = brow[k]; b.y = brow[k + 1];
        // 8 args: (neg_a, A, neg_b, B, c_mod, C, reuse_a, reuse_b)
        c = __builtin_amdgcn_wmma_f32_16x16x4_f32(
                false, a, false, b, (short)0, c, false, false);
    }

#pragma unroll
    for (int v = 0; v < 8; ++v) {
        float x = c[v];
        out[(long long)(rowbase + v + 8 * half) * DFEAT + colbase + m] =
            x > 0.f ? x : 0.f;
    }
}

// ---------------------------------------------------------------- launch
extern "C" void kernel_launch(void* const* d_in, const int* in_sizes, int n_in,
                              void* d_out, int out_size, void* d_ws, size_t ws_size,
                              hipStream_t stream) {
    const float*     h   = (const float*)d_in[0];
    const long long* src = (const long long*)d_in[1];
    const long long* dst = (const long long*)d_in[2];
    const float*     W   = (const float*)d_in[3];
    float*           out = (float*)d_out;
    float*           agg = (float*)d_ws;           // 100000*64*4 = 25.6 MB

    // 1) agg = 0
    const int n_agg = N_NODES * DFEAT;             // 6.4M floats, /4 per thread
    gcn_zero_agg<<<(n_agg / 4 + 255) / 256, 256, 0, stream>>>(agg, n_agg);

    // 2) gather + atomic scatter-add (20M threads, 16 per edge)
    const long long n_scatter = (long long)N_EDGES * 16;
    gcn_scatter_edges<<<(unsigned)(n_scatter / 256), 256, 0, stream>>>(
        h, src, dst, agg);

    // 3) WMMA GEMM + ReLU: 6250 row strips x 4 col tiles (4 waves/block)
    gcn_gemm_relu<<<N_NODES / 16, 128, 0, stream>>>(agg, W, out);
}